// GnnDQNAgent_8890582303024
// MI455X (gfx1250) — compile-verified
//
#include <hip/hip_runtime.h>
#include <hip/hip_bf16.h>

typedef __attribute__((ext_vector_type(2))) float v2f;
typedef __attribute__((ext_vector_type(8))) float v8f;

#define WAVE 32

// ---------------------------------------------------------------------------
// WMMA f32 GEMM: OUT[n,COUT] = A[n,K] @ W[K,COUT] (+bias)(+relu)
// One wave computes one 16x16 output tile with V_WMMA_F32_16X16X4_F32.
// K, COUT compile-time: all loads are base + immediate offset; K-loop fully
// unrolled so the compiler can clause loads and overlap them with WMMAs.
// blockDim.x = 32*(COUT/16): wave w owns column tile w; blockIdx.x = row tile.
// ---------------------------------------------------------------------------
template <int K, int COUT>
__global__ void gemm_wmma(const float* __restrict__ A,
                          const float* __restrict__ W,
                          const float* __restrict__ bias,
                          float* __restrict__ out,
                          int n, int doRelu)
{
    const int wave = threadIdx.x >> 5;
    const int lane = threadIdx.x & 31;
    const int row0 = blockIdx.x * 16;
    const int col0 = wave * 16;

    const int m     = lane & 15;   // A row / B,D col within tile
    const int khalf = lane >> 4;   // K pair {0,1} vs {2,3}

    int aRow = row0 + m;
    if (aRow >= n) aRow = n - 1;   // clamp; garbage rows are store-masked below

    // All per-step accesses are immediate offsets off these two bases.
    const float* aBase = A + (size_t)aRow * K + khalf * 2;
    const float* bBase = W + (size_t)(khalf * 2) * COUT + col0 + m;

    v8f acc = {};
    #pragma unroll
    for (int k = 0; k < K; k += 4) {
        // A 16x4: lanes 0-15 -> K={k,k+1}, lanes 16-31 -> K={k+2,k+3} (b64 load)
        v2f a = *(const v2f*)(aBase + k);
        // B 4x16 mirrored: two b32 loads, COUT apart
        v2f b;
        b.x = bBase[k * COUT];
        b.y = bBase[k * COUT + COUT];
        acc = __builtin_amdgcn_wmma_f32_16x16x4_f32(
                  /*neg_a=*/false, a, /*neg_b=*/false, b,
                  /*c_mod=*/(short)0, acc, /*reuse_a=*/false, /*reuse_b=*/false);
    }

    // C/D 16x16 f32: lane l holds N=l%16; VGPR v holds M = v + 8*(l/16)
    const int nOut = col0 + m;
    #pragma unroll
    for (int v = 0; v < 8; ++v) {
        const int mOut = row0 + v + 8 * khalf;
        if (mOut < n) {
            float val = acc[v];
            if (bias)   val += bias[nOut];
            if (doRelu) val  = fmaxf(val, 0.0f);
            out[(size_t)mOut * COUT + nOut] = val;
        }
    }
}

// ---------------------------------------------------------------------------
// Degree / normalization
// ---------------------------------------------------------------------------
__global__ void deg_init_kernel(float* deg, int n) {
    int i = blockIdx.x * blockDim.x + threadIdx.x;
    if (i < n) deg[i] = 1.0f;                 // self-loop weight
}

__global__ void deg_acc_kernel(const long long* __restrict__ col,
                               const float* __restrict__ w,
                               float* deg, int E) {
    int e = blockIdx.x * blockDim.x + threadIdx.x;
    if (e < E) atomicAdd(&deg[(int)col[e]], w[e]);
}

__global__ void dinv_kernel(float* deg, int n) {
    int i = blockIdx.x * blockDim.x + threadIdx.x;
    if (i < n) {
        float d = deg[i];
        deg[i] = (d > 0.0f) ? rsqrtf(fmaxf(d, 1e-12f)) : 0.0f;
    }
}

__global__ void norm_kernel(const long long* __restrict__ row,
                            const long long* __restrict__ col,
                            const float* __restrict__ w,
                            const float* __restrict__ dis,
                            float* __restrict__ norm, int E) {
    int e = blockIdx.x * blockDim.x + threadIdx.x;
    if (e < E) norm[e] = dis[(int)row[e]] * w[e] * dis[(int)col[e]];
}

// agg[i][c] = dis[i]^2 * xw[i][c] (self-loop contribution; zero-inits agg).
// 2-D grid: blockIdx.x = node, threadIdx.x = feature (no integer division).
__global__ void selfloop_init_kernel(const float* __restrict__ dis,
                                     const float* __restrict__ xw,
                                     float* __restrict__ agg, int C) {
    const int node = blockIdx.x;
    const int f = threadIdx.x;               // blockDim.x == C
    const float d = dis[node];
    const size_t i = (size_t)node * C + f;
    agg[i] = d * d * xw[i];
}

// ---------------------------------------------------------------------------
// Edge scatter: one wave per edge; float4 gather, scalar f32 atomics.
// ---------------------------------------------------------------------------
template <int C>
__global__ void scatter_kernel(const long long* __restrict__ row,
                               const long long* __restrict__ col,
                               const float* __restrict__ norm,
                               const float* __restrict__ xw,
                               float* __restrict__ agg, int E) {
    int e = blockIdx.x * (blockDim.x >> 5) + (threadIdx.x >> 5);
    if (e >= E) return;
    int lane = threadIdx.x & 31;
    int r = (int)row[e];
    int c = (int)col[e];
    float w = norm[e];
    const float* src = xw + (size_t)r * C;
    float* dst = agg + (size_t)c * C;
    __builtin_prefetch(src, 0, 0);           // global_prefetch of gathered row
    #pragma unroll
    for (int f = lane * 4; f < C; f += 128) {
        float4 v = *(const float4*)(src + f);
        atomicAdd(&dst[f + 0], w * v.x);
        atomicAdd(&dst[f + 1], w * v.y);
        atomicAdd(&dst[f + 2], w * v.z);
        atomicAdd(&dst[f + 3], w * v.w);
    }
}

// ---------------------------------------------------------------------------
// bias + LayerNorm + ReLU, one wave per node, wave32 shfl_xor reductions.
// ---------------------------------------------------------------------------
template <int C>
__global__ void ln_relu_kernel(const float* __restrict__ agg,
                               const float* __restrict__ bias,
                               const float* __restrict__ g,
                               const float* __restrict__ be,
                               float* __restrict__ out, int n) {
    int node = blockIdx.x * (blockDim.x >> 5) + (threadIdx.x >> 5);
    if (node >= n) return;
    int lane = threadIdx.x & 31;
    const float* src = agg + (size_t)node * C;
    constexpr int PER = C / 32;              // 4 for C=128, 2 for C=64
    float vals[PER];
    float s = 0.0f;
    #pragma unroll
    for (int j = 0; j < PER; ++j) {
        int f = lane + 32 * j;
        vals[j] = src[f] + bias[f];
        s += vals[j];
    }
    for (int msk = 16; msk > 0; msk >>= 1) s += __shfl_xor(s, msk, 32);
    float mu = s / (float)C;
    float var = 0.0f;
    #pragma unroll
    for (int j = 0; j < PER; ++j) { float d = vals[j] - mu; var += d * d; }
    for (int msk = 16; msk > 0; msk >>= 1) var += __shfl_xor(var, msk, 32);
    float rstd = rsqrtf(var / (float)C + 1e-5f);
    #pragma unroll
    for (int j = 0; j < PER; ++j) {
        int f = lane + 32 * j;
        float y = (vals[j] - mu) * rstd * g[f] + be[f];
        out[(size_t)node * C + f] = fmaxf(y, 0.0f);
    }
}

// ---------------------------------------------------------------------------
// Heads
// ---------------------------------------------------------------------------
__global__ void zero_kernel(float* p, int n) {
    int i = blockIdx.x * blockDim.x + threadIdx.x;
    if (i < n) p[i] = 0.0f;
}

// advantage[node] = dot(hiddenA[node, 0:64], Wa2) + ba2 ; accumulate sum
__global__ void adv_kernel(const float* __restrict__ hidden,
                           const float* __restrict__ Wa2,
                           const float* __restrict__ ba2,
                           float* __restrict__ adv, float* advSum, int n) {
    int node = blockIdx.x * (blockDim.x >> 5) + (threadIdx.x >> 5);
    if (node >= n) return;
    int lane = threadIdx.x & 31;
    const float* src = hidden + (size_t)node * 64;
    float s = src[lane] * Wa2[lane] + src[lane + 32] * Wa2[lane + 32];
    for (int msk = 16; msk > 0; msk >>= 1) s += __shfl_xor(s, msk, 32);
    if (lane == 0) {
        float a = s + ba2[0];
        adv[node] = a;
        atomicAdd(advSum, a);
    }
}

// column sums of h[N,64] -> pool[64]; each block partial-sums 128 nodes
__global__ void pool_kernel(const float* __restrict__ h, float* pool, int n) {
    const int f = threadIdx.x;               // 64 threads = 64 features
    const int start = blockIdx.x * 128;
    float s = 0.0f;
    for (int i = 0; i < 128; ++i) {
        int node = start + i;
        if (node < n) s += h[(size_t)node * 64 + f];
    }
    atomicAdd(&pool[f], s);
}

// value = relu(pool/N @ Wv1 + bv1) @ Wv2 + bv2 ; single block of 64 threads
__global__ void value_kernel(const float* __restrict__ pool,
                             const float* __restrict__ Wv1,
                             const float* __restrict__ bv1,
                             const float* __restrict__ Wv2,
                             const float* __restrict__ bv2,
                             float* value, int n) {
    __shared__ float hp[64];
    __shared__ float red[64];
    int t = threadIdx.x;
    hp[t] = pool[t] / (float)n;
    __syncthreads();
    float acc = bv1[t];
    for (int i = 0; i < 64; ++i) acc += hp[i] * Wv1[i * 64 + t];
    red[t] = fmaxf(acc, 0.0f) * Wv2[t];
    __syncthreads();
    for (int s = 32; s > 0; s >>= 1) {
        if (t < s) red[t] += red[t + s];
        __syncthreads();
    }
    if (t == 0) *value = red[0] + bv2[0];
}

__global__ void q_kernel(const float* __restrict__ adv,
                         const float* __restrict__ value,
                         const float* __restrict__ advSum,
                         float* __restrict__ q, int n) {
    int i = blockIdx.x * blockDim.x + threadIdx.x;
    if (i < n) q[i] = *value + adv[i] - (*advSum) / (float)n;
}

// ---------------------------------------------------------------------------
// Host-side orchestration
// ---------------------------------------------------------------------------
extern "C" void kernel_launch(void* const* d_in, const int* in_sizes, int n_in,
                              void* d_out, int out_size, void* d_ws, size_t ws_size,
                              hipStream_t stream) {
    const int F_IN = 64, HID = 128, EMB = 64;
    const int N = in_sizes[0] / F_IN;
    const int E = in_sizes[2];

    const float*     x   = (const float*)d_in[0];
    const long long* ei  = (const long long*)d_in[1];   // int64 in reference
    const float*     ew  = (const float*)d_in[2];
    const float*     W1  = (const float*)d_in[3];
    const float*     b1  = (const float*)d_in[4];
    const float*     W2  = (const float*)d_in[5];
    const float*     b2  = (const float*)d_in[6];
    const float*     W3  = (const float*)d_in[7];
    const float*     b3  = (const float*)d_in[8];
    const float*     g1  = (const float*)d_in[9];
    const float*     be1 = (const float*)d_in[10];
    const float*     g2  = (const float*)d_in[11];
    const float*     be2 = (const float*)d_in[12];
    const float*     g3  = (const float*)d_in[13];
    const float*     be3 = (const float*)d_in[14];
    const float*     Wv1 = (const float*)d_in[15];
    const float*     bv1 = (const float*)d_in[16];
    const float*     Wv2 = (const float*)d_in[17];
    const float*     bv2 = (const float*)d_in[18];
    const float*     Wa1 = (const float*)d_in[19];
    const float*     ba1 = (const float*)d_in[20];
    const float*     Wa2 = (const float*)d_in[21];
    const float*     ba2 = (const float*)d_in[22];

    const long long* rowIdx = ei;        // edge_index[0,:]
    const long long* colIdx = ei + E;    // edge_index[1,:]

    // workspace layout (floats)
    float* ws   = (float*)d_ws;
    float* dis  = ws;                       size_t off = (size_t)N;
    float* norm = ws + off;                 off += (size_t)E;
    float* B0   = ws + off;                 off += (size_t)N * HID;
    float* B1   = ws + off;                 off += (size_t)N * HID;
    float* B2   = ws + off;                 off += (size_t)N * HID;
    float* smal = ws + off;                 // pool[64], advSum, value
    float* pool   = smal;
    float* advSum = smal + 64;
    float* value  = smal + 65;

    const int TB = 256;
    auto cdiv = [](int a, int b) { return (a + b - 1) / b; };

    // ---- gcn_norm -------------------------------------------------------
    deg_init_kernel<<<cdiv(N, TB), TB, 0, stream>>>(dis, N);
    deg_acc_kernel<<<cdiv(E, TB), TB, 0, stream>>>(colIdx, ew, dis, E);
    dinv_kernel<<<cdiv(N, TB), TB, 0, stream>>>(dis, N);
    norm_kernel<<<cdiv(E, TB), TB, 0, stream>>>(rowIdx, colIdx, ew, dis, norm, E);

    const int rowTiles = cdiv(N, 16);
    const int scatterBlocks = cdiv(E, TB / WAVE);  // 8 edges (waves) per block

    // ---- layer 1: x[N,64] -> B2[N,128] ----------------------------------
    gemm_wmma<64, 128><<<rowTiles, 32 * (128 / 16), 0, stream>>>(x, W1, nullptr, B0, N, 0);
    selfloop_init_kernel<<<N, HID, 0, stream>>>(dis, B0, B1, HID);
    scatter_kernel<128><<<scatterBlocks, TB, 0, stream>>>(rowIdx, colIdx, norm, B0, B1, E);
    ln_relu_kernel<128><<<cdiv(N, TB / WAVE), TB, 0, stream>>>(B1, b1, g1, be1, B2, N);

    // ---- layer 2: B2[N,128] -> B0[N,128] --------------------------------
    gemm_wmma<128, 128><<<rowTiles, 32 * (128 / 16), 0, stream>>>(B2, W2, nullptr, B0, N, 0);
    selfloop_init_kernel<<<N, HID, 0, stream>>>(dis, B0, B1, HID);
    scatter_kernel<128><<<scatterBlocks, TB, 0, stream>>>(rowIdx, colIdx, norm, B0, B1, E);
    ln_relu_kernel<128><<<cdiv(N, TB / WAVE), TB, 0, stream>>>(B1, b2, g2, be2, B0, N);

    // ---- layer 3: B0[N,128] -> B1[N,64] ---------------------------------
    gemm_wmma<128, 64><<<rowTiles, 32 * (64 / 16), 0, stream>>>(B0, W3, nullptr, B1, N, 0);
    selfloop_init_kernel<<<N, EMB, 0, stream>>>(dis, B1, B2, EMB);
    scatter_kernel<64><<<scatterBlocks, TB, 0, stream>>>(rowIdx, colIdx, norm, B1, B2, E);
    ln_relu_kernel<64><<<cdiv(N, TB / WAVE), TB, 0, stream>>>(B2, b3, g3, be3, B1, N);
    // final node embeddings h = B1[N,64]

    // ---- heads ----------------------------------------------------------
    zero_kernel<<<1, 128, 0, stream>>>(smal, 128);   // pool + advSum + value

    // advantage stream: hiddenA = relu(h @ Wa1 + ba1) via WMMA (bias+relu fused)
    gemm_wmma<64, 64><<<rowTiles, 32 * (64 / 16), 0, stream>>>(B1, Wa1, ba1, B0, N, 1);
    adv_kernel<<<cdiv(N, TB / WAVE), TB, 0, stream>>>(B0, Wa2, ba2, B2, advSum, N);

    // value stream: mean-pool then tiny MLP
    pool_kernel<<<cdiv(N, 128), 64, 0, stream>>>(B1, pool, N);
    value_kernel<<<1, 64, 0, stream>>>(pool, Wv1, bv1, Wv2, bv2, value, N);

    // q = value + (adv - mean(adv))
    q_kernel<<<cdiv(N, TB), TB, 0, stream>>>(B2, value, advSum, (float*)d_out, N);
}